// BFPActivation_3770981286287
// MI455X (gfx1250) — compile-verified
//
#include <hip/hip_runtime.h>
#include <cstdint>

// Block-floating-point quantization (shared-exponent, truncated mantissa).
// activations: (64, 256, 56, 56) fp32, blk=32 channels share max exponent,
// mantissa truncated to mantissa_bits relative to shared exponent.
//
// Memory-bound (411 MB traffic, ~17.6 us floor @ 23.3 TB/s). Strategy:
//  - CDNA5 async global->LDS b128 copies (ASYNCcnt) stage each thread's
//    32-channel column in LDS instead of 128 live VGPRs.
//  - two LDS passes: (1) max biased exponent, (2) quantize + NT store.
//  - 128 thr/WG * 32ch * 16B = 64KB LDS -> 5 WGs per 320KB WGP, ~20 waves.

typedef float v4f __attribute__((ext_vector_type(4)));

#define TPB 128
#define BLK 32
#define HW  3136          // 56*56
#define HW4 (HW / 4)      // float4 columns per plane

__global__ __launch_bounds__(TPB)
void bfp_async_kernel(const float* __restrict__ in,
                      float* __restrict__ out,
                      const int* __restrict__ mbits_p,
                      int ncols)
{
    __shared__ v4f tile[BLK][TPB];   // [channel][thread] = 64 KB

    const int tid = threadIdx.x;
    const int col = blockIdx.x * TPB + tid;
    if (col >= ncols) return;

    const int mbits = mbits_p[0];

    // col -> (n * C/BLK + cb, float4 position within 56x56 plane)
    const int    n_cb = col / HW4;
    const int    pos4 = col - n_cb * HW4;
    const size_t base = (size_t)n_cb * ((size_t)BLK * HW) + (size_t)pos4 * 4;

    // ---- Stage 32 channels into LDS with CDNA5 async copies (ASYNCcnt) ----
    const unsigned lds0 = (unsigned)(uintptr_t)(&tile[0][tid]);
    const float*   gp   = in + base;
#pragma unroll
    for (int c = 0; c < BLK; ++c) {
        unsigned la = lds0 + (unsigned)(c * TPB * (int)sizeof(v4f));
        uint64_t ga = (uint64_t)(uintptr_t)(gp + (size_t)c * HW);
        asm volatile("global_load_async_to_lds_b128 %0, %1, off"
                     :: "v"(la), "v"(ga)
                     : "memory");
    }
    asm volatile("s_wait_asynccnt 0x0" ::: "memory");

    // ---- Pass 1: shared exponent = max biased exponent over the block ----
    // frexp exponent e = biased - 126 for normals; exact zero has biased = 0
    // which maps to e = -126 == MIN_EXP, matching the reference for free.
    unsigned emax[4] = {0u, 0u, 0u, 0u};
#pragma unroll
    for (int c = 0; c < BLK; ++c) {
        v4f x = tile[c][tid];
#pragma unroll
        for (int j = 0; j < 4; ++j) {
            unsigned b = (__float_as_uint(x[j]) >> 23) & 0xFFu;
            emax[j] = (b > emax[j]) ? b : emax[j];
        }
    }

    // shift = mantissa_bits - emax_frexp = mbits + 126 - biased_max
    int sh[4];
#pragma unroll
    for (int j = 0; j < 4; ++j)
        sh[j] = mbits + 126 - (int)emax[j];

    // ---- Pass 2: truncate mantissa against shared exponent, NT store ----
    float* op = out + base;
#pragma unroll
    for (int c = 0; c < BLK; ++c) {
        v4f x = tile[c][tid];
        v4f q;
#pragma unroll
        for (int j = 0; j < 4; ++j)
            q[j] = ldexpf(truncf(ldexpf(x[j], sh[j])), -sh[j]);
        __builtin_nontemporal_store(q, (v4f*)(op + (size_t)c * HW));
    }
}

extern "C" void kernel_launch(void* const* d_in, const int* in_sizes, int n_in,
                              void* d_out, int out_size, void* d_ws, size_t ws_size,
                              hipStream_t stream)
{
    const float* in     = (const float*)d_in[0];
    const int*   mbits  = (const int*)d_in[1];
    // d_in[2] = blk (32), baked into the kernel's tiling.
    float*       out    = (float*)d_out;

    const int total = in_sizes[0];            // 64*256*56*56
    const int ncols = total / (BLK * 4);      // float4 columns of 32 channels
    const int grid  = (ncols + TPB - 1) / TPB;

    bfp_async_kernel<<<grid, TPB, 0, stream>>>(in, out, mbits, ncols);
}